// KPInv_76596446757562
// MI455X (gfx1250) — compile-verified
//
#include <hip/hip_runtime.h>
#include <hip/hip_fp16.h>

// ---------------------------------------------------------------------------
// KPInv fused pipeline for MI455X (gfx1250, wave32, WMMA)
//   k1: s_feats@W1 per 16-row tile (V_WMMA_F32_16X16X4_F32) + deterministic
//       per-block partial sums for BatchNorm stats
//   k2: finalize BN scale/shift (folds gamma/beta/mu/var)
//   k3: per 16-point tile: recompute h tile (WMMA) + BN + leaky -> act (LDS),
//       overlapped with geometry (1-NN kernel point + influence),
//       then conv_weights tile 16x960 = act@W2+b2 via WMMA -> f16 LDS,
//       then gather-weighted aggregation over 32 neighbors (coalesced).
// This avoids ever materializing the 192 MB conv_weights tensor in HBM.
// ---------------------------------------------------------------------------

typedef float v2f __attribute__((ext_vector_type(2)));
typedef float v8f __attribute__((ext_vector_type(8)));

#define NPTS   50000
#define CCH    64
#define KPN    15
#define HNB    32
#define SIGMA  0.7f
#define BN_EPS 1e-5f

#define TILES     ((NPTS + 15) / 16)   // 3125 (exact: 3125*16 == 50000)
#define GRID1     ((TILES + 7) / 8)    // 391 blocks, 8 waves (tiles) each
#define SCALE_OFF (GRID1 * 128)        // floats into ws
#define SHIFT_OFF (SCALE_OFF + 64)

static __device__ __forceinline__ v8f wmma_f32_16x16x4(v2f a, v2f b, v8f c) {
  // 8-arg pattern: (neg_a, A, neg_b, B, c_mod, C, reuse_a, reuse_b)
  return __builtin_amdgcn_wmma_f32_16x16x4_f32(false, a, false, b,
                                               (short)0, c, false, false);
}

// --------------------------- kernel 1: GEMM + BN stats ----------------------
__global__ __launch_bounds__(256) void k1_stats(const float* __restrict__ sfeat,
                                                const float* __restrict__ W1,
                                                float* __restrict__ partials) {
  __shared__ float wsum[8][64];
  __shared__ float wssq[8][64];
  const int tid  = threadIdx.x;
  const int wave = tid >> 5;
  const int lane = tid & 31;
  const int half = lane >> 4;
  const int lm16 = lane & 15;
  const int t    = blockIdx.x * 8 + wave;   // 16-row tile index

  v8f acc[4];
  #pragma unroll
  for (int n0 = 0; n0 < 4; ++n0)
    acc[n0] = (v8f){0.f,0.f,0.f,0.f,0.f,0.f,0.f,0.f};

  if (t < TILES) {                          // wave-uniform: EXEC all ones
    const int row = t * 16 + lm16;
    for (int k0 = 0; k0 < CCH; k0 += 4) {
      const int kk = k0 + 2 * half;
      v2f a = *(const v2f*)(sfeat + row * CCH + kk);   // A: M=lane, K pair
      #pragma unroll
      for (int n0 = 0; n0 < 4; ++n0) {
        const int n = n0 * 16 + lm16;
        v2f b;
        b.x = W1[kk * CCH + n];
        b.y = W1[(kk + 1) * CCH + n];
        acc[n0] = wmma_f32_16x16x4(a, b, acc[n0]);
      }
    }
  }

  // per-lane channel sums (channel = n0*16+lm16; halves hold rows r / r+8)
  #pragma unroll
  for (int n0 = 0; n0 < 4; ++n0) {
    float s = 0.f, q = 0.f;
    #pragma unroll
    for (int r = 0; r < 8; ++r) { float v = acc[n0][r]; s += v; q += v * v; }
    s += __shfl_xor(s, 16);
    q += __shfl_xor(q, 16);
    if (half == 0) { wsum[wave][n0 * 16 + lm16] = s; wssq[wave][n0 * 16 + lm16] = q; }
  }
  __syncthreads();

  if (tid < 64) {                           // fixed-order -> deterministic
    float s = 0.f, q = 0.f;
    #pragma unroll
    for (int w = 0; w < 8; ++w) { s += wsum[w][tid]; q += wssq[w][tid]; }
    partials[blockIdx.x * 128 + tid]      = s;
    partials[blockIdx.x * 128 + 64 + tid] = q;
  }
}

// --------------------------- kernel 2: finalize BN --------------------------
__global__ void k2_finalize(const float* __restrict__ partials,
                            const float* __restrict__ gamma,
                            const float* __restrict__ beta,
                            float* __restrict__ ws) {
  const int ch = threadIdx.x;
  if (ch >= 64) return;
  float s = 0.f, q = 0.f;
  for (int b = 0; b < GRID1; ++b) {
    s += partials[b * 128 + ch];
    q += partials[b * 128 + 64 + ch];
  }
  const float inv_n = 1.0f / (float)NPTS;
  const float mu  = s * inv_n;
  const float var = q * inv_n - mu * mu;
  const float sc  = gamma[ch] * rsqrtf(var + BN_EPS);
  ws[SCALE_OFF + ch] = sc;
  ws[SHIFT_OFF + ch] = beta[ch] - mu * sc;
}

// --------------------------- kernel 3: fused main ---------------------------
__global__ __launch_bounds__(256) void k3_main(const float* __restrict__ qpts,
                                               const float* __restrict__ spts,
                                               const float* __restrict__ sfeat,
                                               const int*   __restrict__ nbidx,
                                               const float* __restrict__ kpts,
                                               const float* __restrict__ W1,
                                               const float* __restrict__ W2,
                                               const float* __restrict__ b2,
                                               const float* __restrict__ ws,
                                               float* __restrict__ out) {
  __shared__ float  kp[45];          // 15 x 3 kernel points
  __shared__ float  act[16 * 68];    // BN+leaky activations, padded stride
  __shared__ __half cw[16 * 960];    // conv weights tile (f16 to fit LDS)
  __shared__ int    jIdx[512];
  __shared__ int    nnK[512];
  __shared__ float  nnI[512];

  const int tid  = threadIdx.x;
  const int wave = tid >> 5;
  const int lane = tid & 31;
  const int half = lane >> 4;
  const int lm16 = lane & 15;
  const int m0   = blockIdx.x * 16;

  if (tid < 45) kp[tid] = kpts[tid];
  __syncthreads();

  if (wave < 4) {
    // ---- h tile (cols wave*16..+16) = s_feats@W1, then BN + leaky -> act
    const int n0  = wave * 16;
    const int row = m0 + lm16;
    v8f acc = (v8f){0.f,0.f,0.f,0.f,0.f,0.f,0.f,0.f};
    for (int k0 = 0; k0 < CCH; k0 += 4) {
      const int kk = k0 + 2 * half;
      v2f a = *(const v2f*)(sfeat + row * CCH + kk);
      v2f b;
      b.x = W1[kk * CCH + n0 + lm16];
      b.y = W1[(kk + 1) * CCH + n0 + lm16];
      acc = wmma_f32_16x16x4(a, b, acc);
    }
    const int ch = n0 + lm16;
    const float sc = ws[SCALE_OFF + ch];
    const float sh = ws[SHIFT_OFF + ch];
    #pragma unroll
    for (int r = 0; r < 8; ++r) {
      const int M = r + 8 * half;
      float v = acc[r] * sc + sh;
      v = (v > 0.f) ? v : 0.1f * v;           // leaky_relu(0.1)
      act[M * 68 + ch] = v;
    }
  } else {
    // ---- geometry: 1-NN kernel point + influence for 512 (m,h) pairs
    #pragma unroll
    for (int i = 0; i < 4; ++i) {
      const int p  = (wave - 4) * 128 + i * 32 + lane;   // pair id in tile
      const int lm = p >> 5;
      const int h  = p & 31;
      const int m  = m0 + lm;
      const int j  = nbidx[m * HNB + h];
      const bool valid = ((unsigned)j < (unsigned)NPTS);
      const int jj = valid ? j : 0;
      const float qx = qpts[m * 3 + 0];
      const float qy = qpts[m * 3 + 1];
      const float qz = qpts[m * 3 + 2];
      const float dx = spts[jj * 3 + 0] - qx;
      const float dy = spts[jj * 3 + 1] - qy;
      const float dz = spts[jj * 3 + 2] - qz;
      float best = 3.4e38f; int bk = 0;
      #pragma unroll
      for (int k = 0; k < KPN; ++k) {
        const float ex = dx - kp[k * 3 + 0];
        const float ey = dy - kp[k * 3 + 1];
        const float ez = dz - kp[k * 3 + 2];
        const float sq = ex * ex + ey * ey + ez * ez;
        if (sq < best) { best = sq; bk = k; }   // first-min tie, matches argmin
      }
      float infl = 1.0f - sqrtf(best) * (1.0f / SIGMA);
      infl = (infl > 0.f) ? infl : 0.f;
      if (!valid) infl = 0.f;                   // padding row -> zero weight
      jIdx[p] = jj; nnK[p] = bk; nnI[p] = infl;
    }
  }
  __syncthreads();

  // ---- conv_weights tile: 16x960 = act@W2 + b2, 60 WMMA col-tiles / 8 waves
  for (int t = wave; t < 60; t += 8) {          // wave-uniform loop
    const int n = t * 16 + lm16;
    v8f acc = (v8f){0.f,0.f,0.f,0.f,0.f,0.f,0.f,0.f};
    for (int k0 = 0; k0 < CCH; k0 += 4) {
      const int kk = k0 + 2 * half;
      v2f a = *(const v2f*)(act + lm16 * 68 + kk);
      v2f b;
      b.x = W2[kk * 960 + n];
      b.y = W2[(kk + 1) * 960 + n];
      acc = wmma_f32_16x16x4(a, b, acc);
    }
    const float bias = b2[n];
    #pragma unroll
    for (int r = 0; r < 8; ++r) {
      const int M = r + 8 * half;
      cw[M * 960 + n] = __float2half(acc[r] + bias);
    }
  }
  __syncthreads();

  // ---- aggregation: wave w handles rows w and w+8; lanes = channels
  #pragma unroll
  for (int rr = 0; rr < 2; ++rr) {
    const int lm = wave + rr * 8;
    float a0 = 0.f, a1 = 0.f;
    for (int h = 0; h < HNB; ++h) {
      const int p = lm * HNB + h;
      const float f = nnI[p];                   // uniform across wave
      if (f > 0.f) {
        const int j = jIdx[p];
        const int k = nnK[p];
        const float w0 = __half2float(cw[lm * 960 + k * 64 + lane]) * f;
        const float w1 = __half2float(cw[lm * 960 + k * 64 + lane + 32]) * f;
        a0 = fmaf(sfeat[j * 64 + lane],      w0, a0);   // 128B coalesced
        a1 = fmaf(sfeat[j * 64 + lane + 32], w1, a1);
      }
    }
    out[(m0 + lm) * 64 + lane]      = a0;
    out[(m0 + lm) * 64 + lane + 32] = a1;
  }
}

// ---------------------------------------------------------------------------
extern "C" void kernel_launch(void* const* d_in, const int* in_sizes, int n_in,
                              void* d_out, int out_size, void* d_ws, size_t ws_size,
                              hipStream_t stream) {
  const float* qpts  = (const float*)d_in[0];
  const float* spts  = (const float*)d_in[1];
  const float* sfeat = (const float*)d_in[2];
  const int*   nbidx = (const int*)  d_in[3];
  const float* kpts  = (const float*)d_in[4];
  const float* W1    = (const float*)d_in[5];
  const float* gamma = (const float*)d_in[6];
  const float* beta  = (const float*)d_in[7];
  const float* W2    = (const float*)d_in[8];
  const float* b2    = (const float*)d_in[9];
  float* out = (float*)d_out;
  float* ws  = (float*)d_ws;   // [0,GRID1*128): partials; then scale/shift

  k1_stats<<<GRID1, 256, 0, stream>>>(sfeat, W1, ws);
  k2_finalize<<<1, 64, 0, stream>>>(ws, gamma, beta, ws);
  k3_main<<<TILES, 256, 0, stream>>>(qpts, spts, sfeat, nbidx, kpts,
                                     W1, W2, b2, ws, out);
  (void)in_sizes; (void)n_in; (void)out_size; (void)ws_size;
}